// OpticalSurface_33294586478821
// MI455X (gfx1250) — compile-verified
//
#include <hip/hip_runtime.h>

// CDNA5 / gfx1250: memory-bound streaming kernel.
// 8.4M rays x (sphere intersect + refraction). ~416 MB moved, ~18 us floor
// at 23.3 TB/s HBM. No matrix ops in this workload -> no WMMA; the wins are
// B128 vector memory + non-temporal hints (stream >> 192MB L2) and native
// v_sqrt_f32 / v_rsq_f32 (TRANS pipe) so VALU never approaches the HBM floor.

typedef float v4f __attribute__((ext_vector_type(4)));

#define R_SPH   50.0f
#define R2_SPH  2500.0f
#define AP2     400.0f            // (DIAM/2)^2
#define ETA     (1.0f / 1.5f)     // N1/N2

__device__ __forceinline__ void trace_ray(
    float px, float py, float pz,
    float vx, float vy, float vz,
    float tx, float ty, float tz,
    float& ox, float& oy, float& oz,
    float& Tx, float& Ty, float& Tz,
    float& vd)
{
    // local frame (SCALE = 1)
    const float plx = px - tx, ply = py - ty, plz = pz - tz;
    // sphere center C = (R, 0, 0)
    const float ocx = plx - R_SPH, ocy = ply, ocz = plz;

    const float b    = ocx * vx + ocy * vy + ocz * vz;
    const float c    = ocx * ocx + ocy * ocy + ocz * ocz - R2_SPH;
    const float disc = b * b - c;
    const float sq   = __builtin_amdgcn_sqrtf(fmaxf(disc, 0.0f)); // v_sqrt_f32
    const float t    = -b - sq;

    const float qx = plx + t * vx;
    const float qy = ply + t * vy;
    const float qz = plz + t * vz;
    const float r2 = qy * qy + qz * qz;

    const bool valid = (disc >= 0.0f) && (t > 0.0f) && (r2 <= AP2);

    // outward normal, oriented against incoming ray
    const float invR = 1.0f / R_SPH;
    float nx = (qx - R_SPH) * invR;
    float ny = qy * invR;
    float nz = qz * invR;
    const float vn = vx * nx + vy * ny + vz * nz;
    if (vn > 0.0f) { nx = -nx; ny = -ny; nz = -nz; }

    // masked collision point / normal for blocked rays
    ox = valid ? (qx + tx) : px;
    oy = valid ? (qy + ty) : py;
    oz = valid ? (qz + tz) : pz;
    if (!valid) { nx = -1.0f; ny = 0.0f; nz = 0.0f; }

    // vector Snell's law with clamp past critical angle
    const float cosi = -(vx * nx + vy * ny + vz * nz);
    const float k    = fmaxf(1.0f - ETA * ETA * (1.0f - cosi * cosi), 0.0f);
    const float m    = ETA * cosi - __builtin_amdgcn_sqrtf(k);    // v_sqrt_f32
    const float Ax   = ETA * vx + m * nx;
    const float Ay   = ETA * vy + m * ny;
    const float Az   = ETA * vz + m * nz;
    const float inv  = __builtin_amdgcn_rsqf(Ax * Ax + Ay * Ay + Az * Az); // v_rsq_f32
    Tx = Ax * inv; Ty = Ay * inv; Tz = Az * inv;

    vd = valid ? 1.0f : 0.0f;
}

__global__ __launch_bounds__(256) void OpticalSurface_33294586478821_kernel(
    const float* __restrict__ P,
    const float* __restrict__ V,
    const float* __restrict__ toff,
    float* __restrict__ out,
    int nRays)
{
    const int gid = blockIdx.x * blockDim.x + threadIdx.x;
    const int i0  = gid * 4;                  // 4 rays per thread
    if (i0 >= nRays) return;

    // uniform constants -> scalar loads (KMcnt path)
    const float tx = toff[0];
    const float ty = toff[1];
    const float tz = toff[2];

    float* __restrict__ ptOut = out;
    float* __restrict__ TOut  = out + 3 * (size_t)nRays;
    float* __restrict__ vOut  = out + 6 * (size_t)nRays;

    if (i0 + 4 <= nRays) {
        // ---- fast path: 4 rays, all traffic as B128 non-temporal ----
        // 32-bit offsets (buffers << 4GB) -> saddr + voffset addressing.
        const v4f* __restrict__ P4 = (const v4f*)P;
        const v4f* __restrict__ V4 = (const v4f*)V;
        const int base = gid * 3;             // 3 float4 = 12 floats = 4 rays

        const v4f p0 = __builtin_nontemporal_load(P4 + base + 0);
        const v4f p1 = __builtin_nontemporal_load(P4 + base + 1);
        const v4f p2 = __builtin_nontemporal_load(P4 + base + 2);
        const v4f w0 = __builtin_nontemporal_load(V4 + base + 0);
        const v4f w1 = __builtin_nontemporal_load(V4 + base + 1);
        const v4f w2 = __builtin_nontemporal_load(V4 + base + 2);

        const float px[4] = {p0.x, p0.w, p1.z, p2.y};
        const float py[4] = {p0.y, p1.x, p1.w, p2.z};
        const float pz[4] = {p0.z, p1.y, p2.x, p2.w};
        const float vx[4] = {w0.x, w0.w, w1.z, w2.y};
        const float vy[4] = {w0.y, w1.x, w1.w, w2.z};
        const float vz[4] = {w0.z, w1.y, w2.x, w2.w};

        float ox[4], oy[4], oz[4], Tx[4], Ty[4], Tz[4], vd[4];
#pragma unroll
        for (int r = 0; r < 4; ++r) {
            trace_ray(px[r], py[r], pz[r], vx[r], vy[r], vz[r],
                      tx, ty, tz,
                      ox[r], oy[r], oz[r], Tx[r], Ty[r], Tz[r], vd[r]);
        }

        const v4f o0 = {ox[0], oy[0], oz[0], ox[1]};
        const v4f o1 = {oy[1], oz[1], ox[2], oy[2]};
        const v4f o2 = {oz[2], ox[3], oy[3], oz[3]};
        const v4f t0 = {Tx[0], Ty[0], Tz[0], Tx[1]};
        const v4f t1 = {Ty[1], Tz[1], Tx[2], Ty[2]};
        const v4f t2 = {Tz[2], Tx[3], Ty[3], Tz[3]};
        const v4f vv = {vd[0], vd[1], vd[2], vd[3]};

        __builtin_nontemporal_store(o0, (v4f*)ptOut + base + 0);
        __builtin_nontemporal_store(o1, (v4f*)ptOut + base + 1);
        __builtin_nontemporal_store(o2, (v4f*)ptOut + base + 2);
        __builtin_nontemporal_store(t0, (v4f*)TOut + base + 0);
        __builtin_nontemporal_store(t1, (v4f*)TOut + base + 1);
        __builtin_nontemporal_store(t2, (v4f*)TOut + base + 2);
        __builtin_nontemporal_store(vv, (v4f*)vOut + gid);
    } else {
        // ---- scalar tail (nRays not a multiple of 4) ----
        for (int r = i0; r < nRays; ++r) {
            float ox, oy, oz, Tx, Ty, Tz, vd;
            trace_ray(P[3 * r + 0], P[3 * r + 1], P[3 * r + 2],
                      V[3 * r + 0], V[3 * r + 1], V[3 * r + 2],
                      tx, ty, tz, ox, oy, oz, Tx, Ty, Tz, vd);
            ptOut[3 * r + 0] = ox; ptOut[3 * r + 1] = oy; ptOut[3 * r + 2] = oz;
            TOut[3 * r + 0]  = Tx; TOut[3 * r + 1]  = Ty; TOut[3 * r + 2]  = Tz;
            vOut[r] = vd;
        }
    }
}

extern "C" void kernel_launch(void* const* d_in, const int* in_sizes, int n_in,
                              void* d_out, int out_size, void* d_ws, size_t ws_size,
                              hipStream_t stream) {
    const float* P    = (const float*)d_in[0];
    const float* V    = (const float*)d_in[1];
    const float* toff = (const float*)d_in[2];
    float* out = (float*)d_out;

    const int nRays   = in_sizes[0] / 3;          // P is [N,3]
    const int threads = (nRays + 3) / 4;          // 4 rays per thread
    const int block   = 256;                      // 8 wave32 per block
    const int grid    = (threads + block - 1) / block;

    OpticalSurface_33294586478821_kernel<<<grid, block, 0, stream>>>(
        P, V, toff, out, nRays);
}